// DGALoss_46583215292859
// MI455X (gfx1250) — compile-verified
//
#include <hip/hip_runtime.h>
#include <stdint.h>

// DGA IMU loss for MI455X (gfx1250, wave32).
// Memory-bound reduction over 50MB of f32 input, f64 math to match the f64
// reference. CDNA5-specific path: async global->LDS staging
// (global_load_async_to_lds_b128 / s_wait_asynccnt) for w_hat & a_hat.
// so3_exp uses a small-angle f64 Horner series (error <1e-18 rel for t2<0.04,
// i.e. below f64 rounding) with exact libm fallback, removing ~1M f64 sincos
// calls from the hot path.

#define DT_C    0.005
#define HUBER_C 0.005

constexpr int BLK  = 64;    // threads per block; each thread = 1 group of 32 samples
constexpr int ROWF = 100;   // padded LDS row stride in floats (96 data + 4 pad -> 2-way banks)

__device__ __forceinline__ void so3_exp3(double x, double y, double z, double* R) {
    double t2 = x * x + y * y + z * z;
    double s, c;
    if (t2 < 0.04) {
        // sin(t)/t and (1-cos t)/t^2, Horner in t2; next omitted terms are
        // ~t2^6/13! and ~t2^6/14! -> < 1e-18 relative at t2=0.04.
        s = 1.0 + t2 * (-1.0 / 6.0 + t2 * (1.0 / 120.0 + t2 * (-1.0 / 5040.0
              + t2 * (1.0 / 362880.0 + t2 * (-1.0 / 39916800.0)))));
        c = 0.5 + t2 * (-1.0 / 24.0 + t2 * (1.0 / 720.0 + t2 * (-1.0 / 40320.0
              + t2 * (1.0 / 3628800.0 + t2 * (-1.0 / 479001600.0)))));
    } else {
        double t = sqrt(t2);
        s = sin(t) / t;
        c = (1.0 - cos(t)) / t2;
    }
    double xx = x * x, yy = y * y, zz = z * z;
    double xy = x * y, xz = x * z, yz = y * z;
    R[0] = 1.0 - c * (yy + zz); R[1] = c * xy - s * z;      R[2] = c * xz + s * y;
    R[3] = c * xy + s * z;      R[4] = 1.0 - c * (xx + zz); R[5] = c * yz - s * x;
    R[6] = c * xz - s * y;      R[7] = c * yz + s * x;      R[8] = 1.0 - c * (xx + yy);
}

__device__ __forceinline__ void mm3(const double* A, const double* B, double* C) {
#pragma unroll
    for (int i = 0; i < 3; ++i)
#pragma unroll
        for (int j = 0; j < 3; ++j)
            C[i * 3 + j] = A[i * 3 + 0] * B[0 + j] + A[i * 3 + 1] * B[3 + j] + A[i * 3 + 2] * B[6 + j];
}

// sum_i huber(|log(P^T X)_i| / HUBER)
__device__ __forceinline__ double gyro_res(const double* P, const double* X) {
    double M[9];
#pragma unroll
    for (int i = 0; i < 3; ++i)
#pragma unroll
        for (int j = 0; j < 3; ++j)
            M[i * 3 + j] = P[0 + i] * X[0 + j] + P[3 + i] * X[3 + j] + P[6 + i] * X[6 + j];
    double tr = M[0] + M[4] + M[8];
    double ct = 0.5 * (tr - 1.0);
    ct = fmin(fmax(ct, -1.0 + 1e-10), 1.0 - 1e-10);
    double th = acos(ct);
    double f;
    if (ct > 1.0 - 1e-6) f = 0.5 + th * th * (1.0 / 12.0);
    else                 f = th / (2.0 * sin(th));
    double v0 = f * (M[7] - M[5]);
    double v1 = f * (M[2] - M[6]);
    double v2 = f * (M[3] - M[1]);
    double h = 0.0, d;
    d = fabs(v0) * (1.0 / HUBER_C); h += (d < 1.0) ? 0.5 * d * d : d - 0.5;
    d = fabs(v1) * (1.0 / HUBER_C); h += (d < 1.0) ? 0.5 * d * d : d - 0.5;
    d = fabs(v2) * (1.0 / HUBER_C); h += (d < 1.0) ? 0.5 * d * d : d - 0.5;
    return h;
}

__device__ __forceinline__ double huber1(double d) {
    d = fabs(d);
    return (d < 1.0) ? 0.5 * d * d : d - 0.5;
}

__global__ void dga_init(double* acc) {
    if (threadIdx.x < 4) acc[threadIdx.x] = 0.0;
}

__global__ __launch_bounds__(BLK) void dga_main(
    const float* __restrict__ w_hat, const float* __restrict__ a_hat,
    const float* __restrict__ xs, const float* __restrict__ dv,
    double* __restrict__ acc, int groups_per_batch, int n_groups) {
    __shared__ float sw[BLK * ROWF];
    __shared__ float sa[BLK * ROWF];

    const int tid = threadIdx.x;
    const unsigned long long blk = blockIdx.x;

    // ---- async stage: 2048 samples (6144 floats = 1536 float4) per array ----
    const uint32_t sw_base = (uint32_t)(uintptr_t)(&sw[0]);
    const uint32_t sa_base = (uint32_t)(uintptr_t)(&sa[0]);
    const unsigned long long gw0 = (unsigned long long)(uintptr_t)w_hat + blk * 24576ull;
    const unsigned long long ga0 = (unsigned long long)(uintptr_t)a_hat + blk * 24576ull;
#pragma unroll 1
    for (int it = 0; it < 24; ++it) {
        int lin  = it * BLK + tid;        // float4 index in block chunk [0,1536)
        int row  = lin / 24;              // owning thread row (24 float4 of data / row)
        int col4 = lin - row * 24;
        uint32_t dst = (uint32_t)(row * ROWF + col4 * 4) * 4u;  // bytes, 16B aligned
        unsigned long long srcoff = (unsigned long long)lin * 16ull;
        uint32_t dw = sw_base + dst;
        unsigned long long aw = gw0 + srcoff;
        asm volatile("global_load_async_to_lds_b128 %0, %1, off"
                     :: "v"(dw), "v"(aw) : "memory");
        uint32_t da = sa_base + dst;
        unsigned long long aa = ga0 + srcoff;
        asm volatile("global_load_async_to_lds_b128 %0, %1, off"
                     :: "v"(da), "v"(aa) : "memory");
    }
    asm volatile("s_wait_asynccnt 0" ::: "memory");
    __syncthreads();

    const int g = (int)blk * BLK + tid;
    double s16 = 0.0, s32 = 0.0, a16 = 0.0, a32 = 0.0;

    if (g < n_groups) {
        const int jb = g % groups_per_batch;
        const float* wr = &sw[tid * ROWF];
        const float* ar = &sa[tid * ROWF];

        // ---- gyro: P0 = prod R[0..15], P1 = prod R[16..31] ----
        double P[9], Q[9], Tm[9], P0[9];
        so3_exp3(DT_C * (double)wr[0], DT_C * (double)wr[1], DT_C * (double)wr[2], P);
        for (int k = 1; k < 16; ++k) {
            so3_exp3(DT_C * (double)wr[3 * k], DT_C * (double)wr[3 * k + 1],
                     DT_C * (double)wr[3 * k + 2], Q);
            mm3(P, Q, Tm);
#pragma unroll
            for (int i = 0; i < 9; ++i) P[i] = Tm[i];
        }
#pragma unroll
        for (int i = 0; i < 9; ++i) P0[i] = P[i];
        so3_exp3(DT_C * (double)wr[48], DT_C * (double)wr[49], DT_C * (double)wr[50], P);
        for (int k = 17; k < 32; ++k) {
            so3_exp3(DT_C * (double)wr[3 * k], DT_C * (double)wr[3 * k + 1],
                     DT_C * (double)wr[3 * k + 2], Q);
            mm3(P, Q, Tm);
#pragma unroll
            for (int i = 0; i < 9; ++i) P[i] = Tm[i];
        }

        // ground-truth increments: xs samples 32g and 32g+16
        const float* xsp = xs + (size_t)g * 96;
        double Xa[9], Xb[9];
        so3_exp3((double)xsp[0], (double)xsp[1], (double)xsp[2], Xa);
        so3_exp3((double)xsp[48], (double)xsp[49], (double)xsp[50], Xb);

        if (2 * jb     >= 5) s16 += gyro_res(P0, Xa);
        if (2 * jb + 1 >= 5) s16 += gyro_res(P, Xb);   // P holds P1
        if (jb >= 5) {
            double R32[9], X32[9];
            mm3(P0, P, R32);
            mm3(Xa, Xb, X32);
            s32 += gyro_res(R32, X32);
        }

        // ---- acc: segment sums of 16 ----
        double sax = 0, say = 0, saz = 0, sbx = 0, sby = 0, sbz = 0;
#pragma unroll
        for (int k = 0; k < 16; ++k) {
            sax += (double)ar[3 * k];      say += (double)ar[3 * k + 1];    saz += (double)ar[3 * k + 2];
            sbx += (double)ar[48 + 3 * k]; sby += (double)ar[49 + 3 * k];   sbz += (double)ar[50 + 3 * k];
        }
        sax *= DT_C; say *= DT_C; saz *= DT_C;
        sbx *= DT_C; sby *= DT_C; sbz *= DT_C;

        const float* dvp = dv + (size_t)g * 96;
        double d0x = dvp[0],  d0y = dvp[1],  d0z = dvp[2];
        double d1x = dvp[48], d1y = dvp[49], d1z = dvp[50];

        a16 = huber1(d0x - sax) + huber1(d0y - say) + huber1(d0z - saz)
            + huber1(d1x - sbx) + huber1(d1y - sby) + huber1(d1z - sbz);
        a32 = huber1((d0x + d1x) - (sax + sbx))
            + huber1((d0y + d1y) - (say + sby))
            + huber1((d0z + d1z) - (saz + sbz));
    }

    // ---- wave32 reduction + f64 atomics ----
#pragma unroll
    for (int off = 16; off > 0; off >>= 1) {
        s16 += __shfl_down(s16, off);
        s32 += __shfl_down(s32, off);
        a16 += __shfl_down(a16, off);
        a32 += __shfl_down(a32, off);
    }
    if ((tid & 31) == 0) {
        atomicAdd(&acc[0], s16);
        atomicAdd(&acc[1], s32);
        atomicAdd(&acc[2], a16);
        atomicAdd(&acc[3], a32);
    }
}

__global__ void dga_fin(const double* __restrict__ acc, float* __restrict__ out,
                        int Bq, int T) {
    double c16 = (double)Bq * (double)(T / 16 - 5) * 3.0;   // gyro_16 element count
    double c32 = (double)Bq * (double)(T / 32 - 5) * 3.0;   // gyro_32 element count
    double n16 = (double)Bq * (double)(T / 16) * 3.0;       // acc_16 element count
    double n32 = (double)Bq * (double)(T / 32) * 3.0;       // acc_32 element count
    // W*HUBER^2 = 1e6 * 0.005^2 = 25 ; gyro_32 additionally /2 ; acc losses *10
    double r = 25.0 * acc[0] / c16
             + 12.5 * acc[1] / c32
             + 10.0 * acc[2] / n16
             + 10.0 * acc[3] / n32;
    out[0] = (float)r;
}

extern "C" void kernel_launch(void* const* d_in, const int* in_sizes, int n_in,
                              void* d_out, int out_size, void* d_ws, size_t ws_size,
                              hipStream_t stream) {
    (void)n_in; (void)out_size; (void)ws_size;
    const float* w  = (const float*)d_in[0];
    const float* a  = (const float*)d_in[1];
    const float* xs = (const float*)d_in[2];
    const float* dv = (const float*)d_in[3];
    double* acc = (double*)d_ws;
    float*  out = (float*)d_out;

    long long n_samples = (long long)in_sizes[0] / 3;   // B*T
    int T  = 32768;
    int Bq = (int)(n_samples / T);
    if (Bq < 1) { Bq = 1; T = (int)n_samples; }
    int n_groups = (int)(n_samples / 32);
    int gpb = T / 32;
    int n_blocks = (n_groups + BLK - 1) / BLK;

    dga_init<<<1, 4, 0, stream>>>(acc);
    dga_main<<<n_blocks, BLK, 0, stream>>>(w, a, xs, dv, acc, gpb, n_groups);
    dga_fin<<<1, 1, 0, stream>>>(acc, out, Bq, T);
}